// EmbeddingBagList_3410204033829
// MI455X (gfx1250) — compile-verified
//
#include <hip/hip_runtime.h>
#include <stdint.h>

// EmbeddingBag (mode='sum') over T tables, fp32.
// weights: [T, V, D], indices: [T, NNZ], offsets: [T, B] (sorted, offsets[:,0]==0)
// out: [T, B, D];  bag b sums rows indices[off[b]:off[b+1]] (last bag -> NNZ).
//
// MI455X plan: gather-bound op, weights (13.3 MB) live in L2. One wave32 per
// bag; lane l owns float4 column slice l (D=128 = 32*4). Bag indices are
// staged into LDS with CDNA5 async global->LDS DMA, double-buffered so the
// DMA for chunk c+1 overlaps the row gathers of chunk c (async completions
// are in-order per wave, so s_wait_asynccnt 1 == "oldest chunk landed").

typedef float v4f __attribute__((ext_vector_type(4)));

#define T_TABLES 26
#define V_ROWS   1000
#define D_DIM    128
#define B_BAGS   8192      // 2^13
#define NNZ_PER  409600
#define WAVES_PER_BLOCK 8  // 256 threads = 8 wave32

__global__ __launch_bounds__(256) void
embedding_bag_sum_kernel(const float* __restrict__ weights,
                         const int*   __restrict__ indices,
                         const int*   __restrict__ offsets,
                         float*       __restrict__ out)
{
    // Two staging buffers of 32 indices per wave (double-buffered async DMA).
    __shared__ int s_idx[WAVES_PER_BLOCK][2][32];

    const int lane = threadIdx.x & 31;
    const int wib  = __builtin_amdgcn_readfirstlane((int)(threadIdx.x >> 5));
    const int flat = __builtin_amdgcn_readfirstlane(
        (int)(blockIdx.x * WAVES_PER_BLOCK + (threadIdx.x >> 5)));
    if (flat >= T_TABLES * B_BAGS) return;

    const int t = flat >> 13;           // flat / 8192
    const int b = flat & (B_BAGS - 1);  // flat % 8192

    const int* offT  = offsets + (size_t)t * B_BAGS;
    const int  start = offT[b];
    const int  end   = (b == B_BAGS - 1) ? NNZ_PER : offT[b + 1];

    const int* idxT = indices + (size_t)t * NNZ_PER;
    const v4f* wtT  = (const v4f*)weights + (size_t)t * V_ROWS * (D_DIM / 4);

    // Low 32 bits of a generic LDS pointer == LDS byte offset.
    const uint32_t ldsBuf0 = (uint32_t)(uintptr_t)(&s_idx[wib][0][0]);
    const uint32_t ldsBuf1 = (uint32_t)(uintptr_t)(&s_idx[wib][1][0]);

    const int nChunks = (end - start + 31) >> 5;   // wave-uniform (scalar branchy loop)

    // Issue async DMA of chunk c's (up to 32) indices into buffer c&1.
    auto issue_chunk = [&](int c) {
        const int p0  = start + (c << 5);
        const int rem = (end - p0 < 32) ? (end - p0) : 32;
        if (lane < rem) {
            uint64_t ga = (uint64_t)(uintptr_t)(idxT + p0 + lane);
            uint32_t la = ((c & 1) ? ldsBuf1 : ldsBuf0) + (uint32_t)(lane << 2);
            asm volatile("global_load_async_to_lds_b32 %0, %1, off"
                         :: "v"(la), "v"(ga) : "memory");
        }
    };

    v4f acc = (v4f)0.0f;

    if (nChunks > 0) issue_chunk(0);

    for (int c = 0; c < nChunks; ++c) {
        // Overlap: kick off next chunk's index DMA before consuming this one.
        if (c + 1 < nChunks) {
            issue_chunk(c + 1);
            // In-order completion: <=1 outstanding means chunk c has landed.
            asm volatile("s_wait_asynccnt 0x1" ::: "memory");
        } else {
            asm volatile("s_wait_asynccnt 0x0" ::: "memory");
        }

        const int buf = c & 1;
        const int p0  = start + (c << 5);
        const int rem = (end - p0 < 32) ? (end - p0) : 32;

        // Warm this lane's column slice of its own upcoming row.
        if (lane < rem) {
            const int own = s_idx[wib][buf][lane];
            __builtin_prefetch(&wtT[(size_t)own * (D_DIM / 4) + lane], 0, 3);
        }

        // Gather + accumulate: batches of 8 keep 8 b128 loads in flight.
        int k = 0;
        for (; k + 8 <= rem; k += 8) {
            int id[8];
            v4f v[8];
#pragma unroll
            for (int j = 0; j < 8; ++j) id[j] = s_idx[wib][buf][k + j];
#pragma unroll
            for (int j = 0; j < 8; ++j)
                v[j] = wtT[(size_t)id[j] * (D_DIM / 4) + lane];
#pragma unroll
            for (int j = 0; j < 8; ++j) acc += v[j];
        }
        for (; k < rem; ++k) {
            const int id = s_idx[wib][buf][k];
            acc += wtT[(size_t)id * (D_DIM / 4) + lane];
        }
    }

    // One coalesced 512B store per bag (empty bags write zeros, matching reference).
    v4f* out4 = (v4f*)out + (size_t)flat * (D_DIM / 4);
    out4[lane] = acc;
}

extern "C" void kernel_launch(void* const* d_in, const int* in_sizes, int n_in,
                              void* d_out, int out_size, void* d_ws, size_t ws_size,
                              hipStream_t stream)
{
    (void)in_sizes; (void)n_in; (void)out_size; (void)d_ws; (void)ws_size;

    const float* weights = (const float*)d_in[0];
    const int*   indices = (const int*)d_in[1];
    const int*   offsets = (const int*)d_in[2];
    float*       out     = (float*)d_out;

    const int totalWaves = T_TABLES * B_BAGS;                         // 212992 bags
    dim3 grid((totalWaves + WAVES_PER_BLOCK - 1) / WAVES_PER_BLOCK);  // 26624 blocks
    dim3 block(WAVES_PER_BLOCK * 32);                                 // 256 threads

    hipLaunchKernelGGL(embedding_bag_sum_kernel, grid, block, 0, stream,
                       weights, indices, offsets, out);
}